// NetworkGNN_4544075399263
// MI455X (gfx1250) — compile-verified
//
#include <hip/hip_runtime.h>
#include <hip/hip_bf16.h>
#include <math.h>

// ---------------------------------------------------------------------------
// GCN (3-layer) + mean-pool readouts + classifier head for MI455X (gfx1250).
// Dense GEMMs use V_WMMA_F32_16X16X4_F32 (fp32-exact, wave32).
// Sparse aggregation: wave-per-edge float4 gather + global_atomic_add_f32.
// hw/agg (51 MB each) fit in the 192 MB L2, so gather+atomics stay on-chip.
// ---------------------------------------------------------------------------

#define HID 128
#define OUTD 10

typedef float v2f __attribute__((ext_vector_type(2)));
typedef float v8f __attribute__((ext_vector_type(8)));

__device__ __forceinline__ float eluf(float x) {
    return x > 0.f ? x : (__expf(x) - 1.f);
}

// ---- degree (target side, self-loop added later) --------------------------
__global__ void k_deg(const int* __restrict__ col, float* __restrict__ deg, int E) {
    int i = blockIdx.x * blockDim.x + threadIdx.x;
    if (i < E) atomicAdd(&deg[col[i]], 1.f);
}

// deg -> dinv in place (adds self-loop contribution of +1)
__global__ void k_dinv(float* __restrict__ deg, int n) {
    int i = blockIdx.x * blockDim.x + threadIdx.x;
    if (i < n) {
        float d = deg[i] + 1.f;
        deg[i] = rsqrtf(fmaxf(d, 1.f));
    }
}

// ---- per-graph node counts -> inverse counts ------------------------------
__global__ void k_counts(const int* __restrict__ batch, float* __restrict__ counts, int n) {
    int i = blockIdx.x * blockDim.x + threadIdx.x;
    if (i < n) atomicAdd(&counts[batch[i]], 1.f);
}
__global__ void k_invcnt(float* __restrict__ counts, int g) {
    int i = blockIdx.x * blockDim.x + threadIdx.x;
    if (i < g) counts[i] = 1.f / fmaxf(counts[i], 1.f);
}

// ---------------------------------------------------------------------------
// GEMM: out[M,128] = act(A[M,128] @ W[128,128] + bias[128])
// One wave per 16-row tile; 8 x v8f accumulators cover all 128 columns.
// V_WMMA_F32_16X16X4_F32 layouts:
//   A 16x4 : lanes 0-15 -> M=lid, vgpr0=K0,vgpr1=K1; lanes 16-31 -> K2,K3
//   B 4x16 : vgpr0 holds row K=2*half, vgpr1 row K=2*half+1, col = n0+lid
//   C 16x16: acc[v] = C[m0 + v + 8*half][n0 + lid]
// ---------------------------------------------------------------------------
__global__ void k_gemm128_wmma(const float* __restrict__ A,
                               const float* __restrict__ W,
                               const float* __restrict__ bias,
                               float* __restrict__ out,
                               int M, int apply_elu)
{
    int gwid = (blockIdx.x * blockDim.x + threadIdx.x) >> 5;  // wave id = m-tile
    int lane = threadIdx.x & 31;
    if (gwid * 16 >= M) return;                // wave-uniform: EXEC stays all-1s
    int m0   = gwid * 16;
    int half = lane >> 4;
    int lid  = lane & 15;

    v8f acc[8];
    v8f zero = {0.f, 0.f, 0.f, 0.f, 0.f, 0.f, 0.f, 0.f};
#pragma unroll
    for (int n = 0; n < 8; ++n) acc[n] = zero;

    const float* Arow = A + (size_t)(m0 + lid) * HID;
    for (int k = 0; k < HID; k += 4) {
        int ka = k + 2 * half;
        float2 av = *(const float2*)(Arow + ka);   // 8B aligned (ka even)
        v2f a; a.x = av.x; a.y = av.y;
#pragma unroll
        for (int n = 0; n < 8; ++n) {
            int c = n * 16 + lid;
            v2f b;
            b.x = W[(size_t)ka * HID + c];
            b.y = W[(size_t)(ka + 1) * HID + c];
            acc[n] = __builtin_amdgcn_wmma_f32_16x16x4_f32(
                false, a, false, b, (short)0, acc[n], false, false);
        }
    }

#pragma unroll
    for (int n = 0; n < 8; ++n) {
        int c = n * 16 + lid;
        float bv = bias[c];
#pragma unroll
        for (int v = 0; v < 8; ++v) {
            int m = m0 + v + 8 * half;
            float val = acc[n][v] + bv;
            if (apply_elu) val = eluf(val);
            out[(size_t)m * HID + c] = val;
        }
    }
}

// ---- self-loop term initializes agg: agg[i] = dinv[i]^2 * hw[i] -----------
__global__ void k_selfloop(const float* __restrict__ dinv,
                           const float* __restrict__ hw,
                           float* __restrict__ agg, int n)
{
    int i = blockIdx.x * blockDim.x + threadIdx.x;   // one thread per float4
    if (i >= n * 32) return;
    int node = i >> 5, q = i & 31;
    float s = dinv[node]; s = s * s;
    float4 v = ((const float4*)hw)[(size_t)node * 32 + q];
    float4 o; o.x = s * v.x; o.y = s * v.y; o.z = s * v.z; o.w = s * v.w;
    ((float4*)agg)[(size_t)node * 32 + q] = o;
}

// ---- edge scatter: one wave per edge, lane = float4 chunk of 128 dims -----
__global__ void k_edge_scatter(const int* __restrict__ row,
                               const int* __restrict__ col,
                               const float* __restrict__ dinv,
                               const float* __restrict__ hw,
                               float* __restrict__ agg, int E)
{
    int e    = (blockIdx.x * blockDim.x + threadIdx.x) >> 5;
    int lane = threadIdx.x & 31;
    if (e >= E) return;
    int s = row[e];
    int t = col[e];
    float nrm = dinv[s] * dinv[t];
    float4 v = ((const float4*)(hw + (size_t)s * HID))[lane];
    float* dst = agg + (size_t)t * HID + lane * 4;
    atomicAdd(dst + 0, nrm * v.x);
    atomicAdd(dst + 1, nrm * v.y);
    atomicAdd(dst + 2, nrm * v.z);
    atomicAdd(dst + 3, nrm * v.w);
}

// ---- fused: h = elu(agg + bias); reps[batch] += invcnt * h ----------------
__global__ void k_act_readout(const float* __restrict__ agg,
                              const float* __restrict__ bias,
                              const int* __restrict__ batch,
                              const float* __restrict__ invcnt,
                              float* __restrict__ h,
                              float* __restrict__ reps, int n)
{
    int i = blockIdx.x * blockDim.x + threadIdx.x;
    if (i >= n * 32) return;
    int node = i >> 5, q = i & 31;
    int g = batch[node];
    float ic = invcnt[g];
    float4 v = ((const float4*)agg)[(size_t)node * 32 + q];
    const float* bp = bias + q * 4;
    float4 hv;
    hv.x = eluf(v.x + bp[0]);
    hv.y = eluf(v.y + bp[1]);
    hv.z = eluf(v.z + bp[2]);
    hv.w = eluf(v.w + bp[3]);
    ((float4*)h)[(size_t)node * 32 + q] = hv;
    float* rp = reps + (size_t)g * HID + q * 4;
    atomicAdd(rp + 0, ic * hv.x);
    atomicAdd(rp + 1, ic * hv.y);
    atomicAdd(rp + 2, ic * hv.z);
    atomicAdd(rp + 3, ic * hv.w);
}

// ---- readout only (after lin1, h already has elu applied) -----------------
__global__ void k_readout(const float* __restrict__ h,
                          const int* __restrict__ batch,
                          const float* __restrict__ invcnt,
                          float* __restrict__ reps, int n)
{
    int i = blockIdx.x * blockDim.x + threadIdx.x;
    if (i >= n * 32) return;
    int node = i >> 5, q = i & 31;
    int g = batch[node];
    float ic = invcnt[g];
    float4 v = ((const float4*)h)[(size_t)node * 32 + q];
    float* rp = reps + (size_t)g * HID + q * 4;
    atomicAdd(rp + 0, ic * v.x);
    atomicAdd(rp + 1, ic * v.y);
    atomicAdd(rp + 2, ic * v.z);
    atomicAdd(rp + 3, ic * v.w);
}

// ---- classifier + log_softmax: one thread per graph -----------------------
__global__ void k_cls_logsoftmax(const float* __restrict__ g,
                                 const float* __restrict__ cls_w,  // [128,10]
                                 const float* __restrict__ cls_b,
                                 float* __restrict__ out, int G)
{
    int r = blockIdx.x * blockDim.x + threadIdx.x;
    if (r >= G) return;
    float logits[OUTD];
#pragma unroll
    for (int j = 0; j < OUTD; ++j) logits[j] = cls_b[j];
    const float* grow = g + (size_t)r * HID;
    for (int d = 0; d < HID; ++d) {
        float gv = grow[d];
        const float* wrow = cls_w + (size_t)d * OUTD;
#pragma unroll
        for (int j = 0; j < OUTD; ++j) logits[j] += gv * wrow[j];
    }
    float mx = logits[0];
#pragma unroll
    for (int j = 1; j < OUTD; ++j) mx = fmaxf(mx, logits[j]);
    float sum = 0.f;
#pragma unroll
    for (int j = 0; j < OUTD; ++j) sum += __expf(logits[j] - mx);
    float lse = mx + __logf(sum);
#pragma unroll
    for (int j = 0; j < OUTD; ++j) out[(size_t)r * OUTD + j] = logits[j] - lse;
}

// ---------------------------------------------------------------------------
extern "C" void kernel_launch(void* const* d_in, const int* in_sizes, int n_in,
                              void* d_out, int out_size, void* d_ws, size_t ws_size,
                              hipStream_t stream)
{
    const float* x         = (const float*)d_in[0];   // [N,128]
    const float* lin1_w    = (const float*)d_in[1];   // [128,128]
    const float* lin1_b    = (const float*)d_in[2];   // [128]
    const float* gcn_w     = (const float*)d_in[3];   // [3,128,128]
    const float* gcn_b     = (const float*)d_in[4];   // [3,128]
    const float* lin_out_w = (const float*)d_in[5];   // [128,128]
    const float* lin_out_b = (const float*)d_in[6];   // [128]
    const float* cls_w     = (const float*)d_in[7];   // [128,10]
    const float* cls_b     = (const float*)d_in[8];   // [10]
    const int*   edge_idx  = (const int*)d_in[9];     // [2,E]
    const int*   batch     = (const int*)d_in[10];    // [N]

    const int N = in_sizes[0] / HID;        // 100000 (multiple of 16)
    const int E = in_sizes[9] / 2;          // 1600000
    const int G = out_size / OUTD;          // 512

    const int* erow = edge_idx;       // source (message origin)
    const int* ecol = edge_idx + E;   // target (aggregation index)

    // workspace layout (floats)
    float* ws     = (float*)d_ws;
    float* h      = ws;                         // N*128
    float* hw     = h   + (size_t)N * HID;      // N*128
    float* agg    = hw  + (size_t)N * HID;      // N*128
    float* dinv   = agg + (size_t)N * HID;      // N   (deg, then rsqrt in place)
    float* invcnt = dinv + N;                   // G   (counts, then 1/count)
    float* zbias  = invcnt + G;                 // HID (all zeros)
    float* reps   = zbias + HID;                // G*128
    float* gbuf   = reps + (size_t)G * HID;     // G*128
    float* out    = (float*)d_out;

    const int TB = 256;
    int nbN   = (N + TB - 1) / TB;
    int nbE   = (E + TB - 1) / TB;
    int nbN4  = (N * 32 + TB - 1) / TB;          // float4-granular node kernels
    int nbE32 = (int)(((long long)E * 32 + TB - 1) / TB);  // wave-per-edge
    int nbG   = (G + TB - 1) / TB;

    // --- init + normalization precompute ---
    hipMemsetAsync(dinv,   0, sizeof(float) * N, stream);
    hipMemsetAsync(invcnt, 0, sizeof(float) * G, stream);
    hipMemsetAsync(zbias,  0, sizeof(float) * HID, stream);
    hipMemsetAsync(reps,   0, sizeof(float) * (size_t)G * HID, stream);

    k_deg<<<nbE, TB, 0, stream>>>(ecol, dinv, E);
    k_dinv<<<nbN, TB, 0, stream>>>(dinv, N);
    k_counts<<<nbN, TB, 0, stream>>>(batch, invcnt, N);
    k_invcnt<<<nbG, TB, 0, stream>>>(invcnt, G);

    // --- h = elu(x @ lin1_w + lin1_b); reps += readout(h) ---
    int gemmBlocksN = (N * 2 + TB - 1) / TB;     // N/16 waves, 32 thr each
    k_gemm128_wmma<<<gemmBlocksN, TB, 0, stream>>>(x, lin1_w, lin1_b, h, N, 1);
    k_readout<<<nbN4, TB, 0, stream>>>(h, batch, invcnt, reps, N);

    // --- 3 GCN layers ---
    for (int layer = 0; layer < 3; ++layer) {
        const float* W = gcn_w + (size_t)layer * HID * HID;
        const float* b = gcn_b + (size_t)layer * HID;
        // hw = h @ W (no bias here; gcn bias applied post-aggregation)
        k_gemm128_wmma<<<gemmBlocksN, TB, 0, stream>>>(h, W, zbias, hw, N, 0);
        k_selfloop<<<nbN4, TB, 0, stream>>>(dinv, hw, agg, N);
        k_edge_scatter<<<nbE32, TB, 0, stream>>>(erow, ecol, dinv, hw, agg, E);
        k_act_readout<<<nbN4, TB, 0, stream>>>(agg, b, batch, invcnt, h, reps, N);
    }

    // --- g = elu(reps @ lin_out_w + lin_out_b) ---
    int gemmBlocksG = (G * 2 + TB - 1) / TB;
    k_gemm128_wmma<<<gemmBlocksG, TB, 0, stream>>>(reps, lin_out_w, lin_out_b, gbuf, G, 1);

    // --- logits + log_softmax ---
    k_cls_logsoftmax<<<nbG, TB, 0, stream>>>(gbuf, cls_w, cls_b, out, G);
}